// TargetHead_52561809768760
// MI455X (gfx1250) — compile-verified
//
#include <hip/hip_runtime.h>

// ---------------------------------------------------------------------------
// TargetHead for MI455X (gfx1250, wave32).
// Dominant cost: streaming entity_encodings (102.4 MB) once -> ~4.4us at
// 23.3 TB/s. keys GEMM is done with v_wmma_f32_16x16x4_f32 and fully fused
// with sigmoid/pow/sum/argmax so keys never touch HBM.
// ---------------------------------------------------------------------------

typedef float v2f __attribute__((ext_vector_type(2)));
typedef float v8f __attribute__((ext_vector_type(8)));

#define TEMP_INV 1.25f            // 1/0.8
#define LN_EPS   1e-5f
#define F32_TINY 1.1754943508222875e-38f

__device__ __forceinline__ float sigmoidf(float x) { return 1.0f / (1.0f + expf(-x)); }
__device__ __forceinline__ unsigned rotl32(unsigned x, unsigned n) { return (x << n) | (x >> (32u - n)); }

// JAX threefry2x32 with key = PRNGKey(42) = (0, 42); returns out0 or out1.
__device__ unsigned threefry2x32_42(unsigned x0, unsigned x1, int take_hi) {
  const unsigned ks0 = 0u, ks1 = 42u, ks2 = 0u ^ 42u ^ 0x1BD11BDAu;
  x0 += ks0; x1 += ks1;
#define TF_R4(r0, r1, r2, r3)                              \
  x0 += x1; x1 = rotl32(x1, r0); x1 ^= x0;                 \
  x0 += x1; x1 = rotl32(x1, r1); x1 ^= x0;                 \
  x0 += x1; x1 = rotl32(x1, r2); x1 ^= x0;                 \
  x0 += x1; x1 = rotl32(x1, r3); x1 ^= x0;
  TF_R4(13, 15, 26, 6)  x0 += ks1; x1 += ks2 + 1u;
  TF_R4(17, 29, 16, 24) x0 += ks2; x1 += ks0 + 2u;
  TF_R4(13, 15, 26, 6)  x0 += ks0; x1 += ks1 + 3u;
  TF_R4(17, 29, 16, 24) x0 += ks1; x1 += ks2 + 4u;
  TF_R4(13, 15, 26, 6)  x0 += ks2; x1 += ks0 + 5u;
#undef TF_R4
  return take_hi ? x1 : x0;
}

// ---------------------------------------------------------------------------
// Kernel 1: tiny MLP + LSTM-cell -> query[32] in workspace; reset accumulators.
// One block of 256 threads (8 waves). Threads 0..31 form wave 0 (wave32), so
// the 32-wide layernorm reductions are pure lane shuffles.
// ---------------------------------------------------------------------------
__global__ void th_k1_query(const float* __restrict__ are,
                            const float* __restrict__ W0, const float* __restrict__ b0,
                            const float* __restrict__ W1, const float* __restrict__ b1,
                            const float* __restrict__ Wi0, const float* __restrict__ bi0,
                            const float* __restrict__ Wi1, const float* __restrict__ bi1,
                            const float* __restrict__ Wo, const float* __restrict__ bo,
                            const float* __restrict__ ln_g, const float* __restrict__ ln_b,
                            float* __restrict__ ws_q, float* __restrict__ ws_sum,
                            unsigned long long* __restrict__ ws_amax) {
  __shared__ float h256[256];
  __shared__ float x32[32];
  const int t = threadIdx.x;

  // Linear(1024 -> 256) + ReLU, one output per thread.
  {
    const float4* w = (const float4*)(W0 + (size_t)t * 1024);
    const float4* a = (const float4*)are;
    float acc = b0[t];
#pragma unroll 4
    for (int k = 0; k < 256; ++k) {
      float4 wv = w[k], av = a[k];
      acc += wv.x * av.x + wv.y * av.y + wv.z * av.z + wv.w * av.w;
    }
    h256[t] = fmaxf(acc, 0.0f);
  }
  __syncthreads();

  // Linear(256 -> 32) + ReLU.  x = concat(h32, zeros(32)) -> only h32 matters
  // for the 64-wide gate matvecs (upper half of x is zero).
  if (t < 32) {
    float acc = b1[t];
    const float* w = W1 + t * 256;
    for (int k = 0; k < 256; ++k) acc += h256[k] * w[k];
    x32[t] = fmaxf(acc, 0.0f);
  }
  __syncthreads();

  if (t < 32) {
    float si = bi0[t], ti = bi1[t], so = bo[t];
    const float* wi0 = Wi0 + t * 64;
    const float* wi1 = Wi1 + t * 64;
    const float* wo  = Wo  + t * 64;
    for (int k = 0; k < 32; ++k) {
      float xv = x32[k];
      si += xv * wi0[k];
      ti += xv * wi1[k];
      so += xv * wo[k];
    }
    float rem = sigmoidf(si) * tanhf(ti);  // remember gate (forget*c_prev == 0)
    float og  = sigmoidf(so);              // output gate

    const float g = ln_g[t], b = ln_b[t];
    auto lnorm = [&](float v) -> float {
      float m = v;
      for (int o = 16; o; o >>= 1) m += __shfl_xor(m, o, 32);
      m *= (1.0f / 32.0f);
      float d = v - m;
      float var = d * d;
      for (int o = 16; o; o >>= 1) var += __shfl_xor(var, o, 32);
      var *= (1.0f / 32.0f);
      return d * rsqrtf(var + LN_EPS) * g + b;
    };
    float new_h = lnorm(rem);
    ws_q[t] = tanhf(new_h) * lnorm(og);
  }
  if (t == 0) { *ws_sum = 0.0f; *ws_amax = 0ull; }
}

// ---------------------------------------------------------------------------
// Kernel 2: fused keys-GEMM (WMMA f32 16x16x4) + query dot + sigmoid + ^1.25
// + global-sum + Gumbel argmax. One wave per 16-entity tile, 8 waves/block.
// ---------------------------------------------------------------------------
__global__ void th_k2_logits(const float* __restrict__ ee,
                             const float* __restrict__ Wk, const float* __restrict__ bk,
                             const float* __restrict__ ws_q,
                             float* __restrict__ out_vec,
                             float* __restrict__ ws_sum,
                             unsigned long long* __restrict__ ws_amax,
                             int E, int halfE, int ntiles) {
  __shared__ float ssum[8];
  __shared__ unsigned long long smax[8];
  const int lane = threadIdx.x & 31;
  const int wid  = threadIdx.x >> 5;
  const int tile = blockIdx.x * 8 + wid;

  float wave_sum = 0.0f;
  unsigned long long wave_max = 0ull;

  if (tile < ntiles) {  // wave-uniform: WMMA executes with full EXEC
    const int nl = lane & 15;
    const int kh = (lane >> 4) << 1;  // 0 for lanes 0-15 (K=k,k+1), 2 for 16-31 (K=k+2,k+3)
    const int rowbase = tile << 4;
    int arow = rowbase + nl;
    if (arow >= E) arow = E - 1;  // clamp (guarded out below)

    const float* ap  = ee + (size_t)arow * 256 + kh;   // A[M=nl][k+kh .. k+kh+1]
    const float* bp0 = Wk + (size_t)nl * 256 + kh;     // B tile0: B[k][N=nl]   = Wk[nl][k]
    const float* bp1 = Wk + (size_t)(nl + 16) * 256 + kh;  // B tile1: N = nl+16

    v8f c0 = {};
    v8f c1 = {};
#pragma unroll 4
    for (int k = 0; k < 256; k += 4) {
      v2f a  = *(const v2f*)(ap + k);
      v2f bb0 = *(const v2f*)(bp0 + k);
      v2f bb1 = *(const v2f*)(bp1 + k);
      c0 = __builtin_amdgcn_wmma_f32_16x16x4_f32(false, a, false, bb0, (short)0, c0, false, false);
      c1 = __builtin_amdgcn_wmma_f32_16x16x4_f32(false, a, false, bb1, (short)0, c1, false, false);
    }

    // keys[m][n] = c[.] + bk[n];  sim[m] = sum_n keys[m][n] * query[n]
    const float q0 = ws_q[nl], q1 = ws_q[nl + 16];
    const float bk0 = bk[nl], bk1 = bk[nl + 16];
    float part[8];
#pragma unroll
    for (int r = 0; r < 8; ++r)
      part[r] = (c0[r] + bk0) * q0 + (c1[r] + bk1) * q1;
    // reduce over N within each 16-lane half (C layout: lanes 0-15 -> M=r, 16-31 -> M=8+r)
#pragma unroll
    for (int r = 0; r < 8; ++r)
      for (int o = 1; o < 16; o <<= 1) part[r] += __shfl_xor(part[r], o, 32);

    float sim = part[0];
#pragma unroll
    for (int r = 1; r < 8; ++r)
      if (nl == r) sim = part[r];

    const int row = rowbase + ((lane >> 4) << 3) + nl;  // lanes nl<8 own one row each
    float vec = 0.0f;
    unsigned long long packed = 0ull;
    if (nl < 8 && row < E) {
      float logit = 1.0f / (1.0f + expf(-sim));
      float lg = logf(logit);            // <= 0; -inf when sigmoid underflows -> vec 0 (nan_to_num)
      vec = expf(TEMP_INV * lg);         // logit^(1/T)
      out_vec[row] = vec;                // unnormalized; k3 divides by the sum

      // jax.random.categorical(key(42)): argmax(log p + gumbel). log-normalizer is
      // a constant shift -> use TEMP_INV*lg directly.
      unsigned p = (row < halfE) ? (unsigned)row : (unsigned)(row - halfE);
      unsigned bits = threefry2x32_42(p, p + (unsigned)halfE, row >= halfE);
      float uf = __uint_as_float((bits >> 9) | 0x3f800000u) - 1.0f;  // [0,1)
      uf = uf * (1.0f - F32_TINY) + F32_TINY;                        // [tiny,1)
      float gum = -logf(-logf(uf));
      float score = TEMP_INV * lg + gum;
      unsigned fb = __float_as_uint(score);
      unsigned ob = (fb & 0x80000000u) ? ~fb : (fb | 0x80000000u);   // order-preserving map
      packed = ((unsigned long long)ob << 32) | (unsigned)(~(unsigned)row);  // ties -> lowest row
    }
    // full-wave sum of vec
    for (int o = 16; o; o >>= 1) vec += __shfl_xor(vec, o, 32);
    wave_sum = vec;
    // full-wave max of packed (64-bit via two 32-bit shuffles)
    for (int o = 16; o; o >>= 1) {
      unsigned lo = __shfl_xor((unsigned)(packed & 0xffffffffu), o, 32);
      unsigned hi = __shfl_xor((unsigned)(packed >> 32), o, 32);
      unsigned long long other = ((unsigned long long)hi << 32) | lo;
      if (other > packed) packed = other;
    }
    wave_max = packed;
  }

  if (lane == 0) { ssum[wid] = wave_sum; smax[wid] = wave_max; }
  __syncthreads();
  if (threadIdx.x == 0) {
    float s = 0.0f;
    unsigned long long m = 0ull;
    for (int w = 0; w < 8; ++w) {
      s += ssum[w];
      if (smax[w] > m) m = smax[w];
    }
    atomicAdd(ws_sum, s);
    atomicMax(ws_amax, m);
  }
}

// ---------------------------------------------------------------------------
// Kernel 3: normalize unit_logits and scatter targeted_unit.
// ---------------------------------------------------------------------------
__global__ void th_k3_finalize(float* __restrict__ out_logits,
                               float* __restrict__ out_target,
                               const unsigned char* __restrict__ entity_mask,
                               const float* __restrict__ ws_sum,
                               const unsigned long long* __restrict__ ws_amax,
                               int E) {
  const int i = blockIdx.x * blockDim.x + threadIdx.x;
  if (i >= E) return;
  const float s = *ws_sum;
  const float inv = (s != 0.0f) ? (1.0f / s) : 1.0f;
  out_logits[i] *= inv;
  const unsigned pick = ~(unsigned)(*ws_amax & 0xffffffffull);
  float tv = 0.0f;
  if ((unsigned)i == pick) tv = entity_mask[pick] ? 0.0f : 1.0f;  // em = !mask
  out_target[i] = tv;
}

// ---------------------------------------------------------------------------
extern "C" void kernel_launch(void* const* d_in, const int* in_sizes, int n_in,
                              void* d_out, int out_size, void* d_ws, size_t ws_size,
                              hipStream_t stream) {
  const unsigned char* entity_mask = (const unsigned char*)d_in[1];  // bool -> 1 byte
  const float* ee  = (const float*)d_in[2];
  const float* are = (const float*)d_in[3];
  const float* Wk  = (const float*)d_in[5];
  const float* bk  = (const float*)d_in[6];
  const float* W0  = (const float*)d_in[7];
  const float* b0  = (const float*)d_in[8];
  const float* W1  = (const float*)d_in[9];
  const float* b1  = (const float*)d_in[10];
  const float* Wi0 = (const float*)d_in[13];
  const float* bi0 = (const float*)d_in[14];
  const float* Wi1 = (const float*)d_in[15];
  const float* bi1 = (const float*)d_in[16];
  const float* Wo  = (const float*)d_in[17];
  const float* bo  = (const float*)d_in[18];
  const float* lng = (const float*)d_in[19];
  const float* lnb = (const float*)d_in[20];

  const int E = in_sizes[1];

  float* wsf = (float*)d_ws;
  float* ws_q = wsf;                                                      // 32 floats
  float* ws_sum = wsf + 32;                                               // 1 float
  unsigned long long* ws_amax = (unsigned long long*)((char*)d_ws + 136); // 8B aligned

  float* out_logits = (float*)d_out;
  float* out_target = out_logits + E;

  th_k1_query<<<1, 256, 0, stream>>>(are, W0, b0, W1, b1, Wi0, bi0, Wi1, bi1,
                                     Wo, bo, lng, lnb, ws_q, ws_sum, ws_amax);

  const int ntiles = (E + 15) / 16;
  const int nblocks = (ntiles + 7) / 8;  // 8 waves/block, one 16-row tile per wave
  th_k2_logits<<<nblocks, 256, 0, stream>>>(ee, Wk, bk, ws_q, out_logits,
                                            ws_sum, ws_amax, E, E / 2, ntiles);

  th_k3_finalize<<<(E + 255) / 256, 256, 0, stream>>>(out_logits, out_target,
                                                      entity_mask, ws_sum, ws_amax, E);
}